// GraphAttentionLayer_67551245632208
// MI455X (gfx1250) — compile-verified
//
#include <hip/hip_runtime.h>
#include <hip/hip_bf16.h>

#define N 4096
#define IN_DIM 512
#define HEADS 8
#define HID 64
#define OUTF (HEADS * HID) /* 512 */
#define SLOPE 0.2f

typedef __attribute__((ext_vector_type(2))) float v2f;
typedef __attribute__((ext_vector_type(8))) float v8f;
typedef int v4i_vs __attribute__((vector_size(4 * sizeof(int))));
typedef __attribute__((address_space(1))) v4i_vs glb_v4i;
typedef __attribute__((address_space(3))) v4i_vs lds_v4i;

__device__ __forceinline__ v8f wmma_f32(v2f a, v2f b, v8f c) {
  // D(16x16,f32) = A(16x4,f32) * B(4x16,f32) + C
  return __builtin_amdgcn_wmma_f32_16x16x4_f32(
      /*neg_a=*/false, a, /*neg_b=*/false, b,
      /*c_mod=*/(short)0, c, /*reuse_a=*/false, /*reuse_b=*/false);
}

// Async global -> LDS 16-byte copy (gfx1250 GLOBAL_LOAD_ASYNC_TO_LDS_B128,
// tracked by ASYNCcnt). Per-lane addresses; writes LDS directly, no VGPR data.
__device__ __forceinline__ void async_cp16(const void* gsrc, void* ldst) {
  __builtin_amdgcn_global_load_async_to_lds_b128(
      (glb_v4i*)gsrc, (lds_v4i*)ldst, /*offset=*/0, /*cpol=*/0);
}
__device__ __forceinline__ void wait_async_le4() {
  asm volatile("s_wait_asynccnt 0x4" ::: "memory");
}
__device__ __forceinline__ void wait_async_0() {
  asm volatile("s_wait_asynccnt 0x0" ::: "memory");
}

// ---------------------------------------------------------------------------
// Kernel 1: g = h @ W  (4096x512 @ 512x512 fp32, WMMA f32 16x16x4)
// 256 thr (8 waves) per 64x64 tile. Double-buffered async LDS staging:
// 4 async b128 instr / wave / K-panel -> s_wait_asynccnt 4 overlaps panels.
// ---------------------------------------------------------------------------
#define SH_STR 36  // 144B row stride: 16B aligned, 36r mod 64 distinct (no conflicts)
#define SW_STR 72  // 288B row stride: 16B aligned, half-waves 16 banks apart

__global__ __launch_bounds__(256) void gemm_g_kernel(
    const float* __restrict__ h, const float* __restrict__ W,
    float* __restrict__ g) {
  __shared__ float sh[2][64][SH_STR];  // h tiles 64x32
  __shared__ float sw[2][32][SW_STR];  // W tiles 32x64

  const int tid  = threadIdx.x;
  const int lane = tid & 31;
  const int wave = tid >> 5;
  const int i0 = blockIdx.x * 64;
  const int j0 = blockIdx.y * 64;
  const int rt = wave >> 1;        // row tile 0..3
  const int ct = (wave & 1) * 2;   // col tiles ct, ct+1

  const int r    = lane & 15;
  const int kb2  = (lane >> 4) << 1;  // K offset 0 or 2
  const int half = lane >> 4;

  v8f c0 = {}; v8f c1 = {};

  // stage K-panel into buffer b: 2 b128 for h tile + 2 b128 for W tile
  auto stage = [&](int k0, int b) {
    #pragma unroll
    for (int e = 0; e < 2; ++e) {
      int ch = e * 256 + tid;               // 512 chunks of 4 floats (64x32)
      int rr = ch >> 3, cc = (ch & 7) << 2;
      async_cp16(&h[(size_t)(i0 + rr) * IN_DIM + k0 + cc], &sh[b][rr][cc]);
    }
    #pragma unroll
    for (int e = 0; e < 2; ++e) {
      int ch = e * 256 + tid;               // 512 chunks of 4 floats (32x64)
      int rr = ch >> 4, cc = (ch & 15) << 2;
      async_cp16(&W[(size_t)(k0 + rr) * OUTF + j0 + cc], &sw[b][rr][cc]);
    }
  };

  stage(0, 0);
  const int NP = IN_DIM / 32;  // 16 K-panels
  for (int p = 0; p < NP; ++p) {
    const int b = p & 1;
    if (p + 1 < NP) { stage((p + 1) * 32, b ^ 1); wait_async_le4(); }
    else            { wait_async_0(); }
    __syncthreads();

    #pragma unroll
    for (int k = 0; k < 32; k += 4) {
      v2f a = *(const v2f*)&sh[b][rt * 16 + r][k + kb2];  // contiguous 8B
      v2f b0, b1;
      b0.x = sw[b][k + kb2][ct * 16 + r];
      b0.y = sw[b][k + kb2 + 1][ct * 16 + r];
      b1.x = sw[b][k + kb2][(ct + 1) * 16 + r];
      b1.y = sw[b][k + kb2 + 1][(ct + 1) * 16 + r];
      c0 = wmma_f32(a, b0, c0);
      c1 = wmma_f32(a, b1, c1);
    }
    __syncthreads();
  }

  #pragma unroll
  for (int v = 0; v < 8; ++v) {  // C layout: row = v + 8*half, col = lane&15
    int row = i0 + rt * 16 + v + 8 * half;
    g[(size_t)row * OUTF + j0 + ct * 16 + r]       = c0[v];
    g[(size_t)row * OUTF + j0 + (ct + 1) * 16 + r] = c1[v];
  }
}

// ---------------------------------------------------------------------------
// Kernel 2: el[i,h] = g[i,h,:].a_l ; er[i,h] = g[i,h,:].a_r
// ---------------------------------------------------------------------------
__global__ __launch_bounds__(256) void elr_kernel(
    const float* __restrict__ g, const float* __restrict__ a,
    float* __restrict__ el, float* __restrict__ er) {
  const int lane = threadIdx.x & 31;
  const int wave = threadIdx.x >> 5;
  const int i = blockIdx.x * 8 + wave;

  #pragma unroll
  for (int hh = 0; hh < HEADS; ++hh) {
    float g0 = g[(size_t)i * OUTF + hh * HID + lane];
    float g1 = g[(size_t)i * OUTF + hh * HID + lane + 32];
    float sl = g0 * a[lane]       + g1 * a[lane + 32];
    float sr = g0 * a[HID + lane] + g1 * a[HID + lane + 32];
    #pragma unroll
    for (int off = 16; off > 0; off >>= 1) {
      sl += __shfl_xor(sl, off, 32);
      sr += __shfl_xor(sr, off, 32);
    }
    if (lane == 0) {
      el[i * HEADS + hh] = sl;
      er[i * HEADS + hh] = sr;
    }
  }
}

// ---------------------------------------------------------------------------
// Kernel 3: masked-softmax row stats (max pass, then exp-sum pass).
// ---------------------------------------------------------------------------
__global__ __launch_bounds__(256) void stats_kernel(
    const unsigned char* __restrict__ adj, const float* __restrict__ el,
    const float* __restrict__ er, float* __restrict__ mrow,
    float* __restrict__ lrow) {
  const int lane = threadIdx.x & 31;
  const int wave = threadIdx.x >> 5;
  const int i = blockIdx.x * 8 + wave;

  float eli[HEADS];
  #pragma unroll
  for (int hh = 0; hh < HEADS; ++hh) eli[hh] = el[i * HEADS + hh];

  float m[HEADS];
  #pragma unroll
  for (int hh = 0; hh < HEADS; ++hh) m[hh] = -1e30f;

  for (int j = lane; j < N; j += 32) {
    if (adj[(size_t)i * N + j]) {
      #pragma unroll
      for (int hh = 0; hh < HEADS; ++hh) {
        float e = eli[hh] + er[j * HEADS + hh];
        e = e >= 0.f ? e : SLOPE * e;
        m[hh] = fmaxf(m[hh], e);
      }
    }
  }
  #pragma unroll
  for (int hh = 0; hh < HEADS; ++hh) {
    #pragma unroll
    for (int off = 16; off > 0; off >>= 1)
      m[hh] = fmaxf(m[hh], __shfl_xor(m[hh], off, 32));
  }
  float l[HEADS];
  #pragma unroll
  for (int hh = 0; hh < HEADS; ++hh) l[hh] = 0.f;

  for (int j = lane; j < N; j += 32) {
    if (adj[(size_t)i * N + j]) {
      #pragma unroll
      for (int hh = 0; hh < HEADS; ++hh) {
        float e = eli[hh] + er[j * HEADS + hh];
        e = e >= 0.f ? e : SLOPE * e;
        l[hh] += __expf(e - m[hh]);
      }
    }
  }
  #pragma unroll
  for (int hh = 0; hh < HEADS; ++hh) {
    #pragma unroll
    for (int off = 16; off > 0; off >>= 1)
      l[hh] += __shfl_xor(l[hh], off, 32);
    if (lane == 0) {
      mrow[i * HEADS + hh] = m[hh];
      lrow[i * HEADS + hh] = 1.f / l[hh];  // diagonal guarantees l > 0
    }
  }
}

// ---------------------------------------------------------------------------
// Kernel 4: out[i,h,:] = sum_j p[i,j,h] * g[j,h,:]  via WMMA f32 16x16x4.
// Block = 8 waves (128 rows x 1 head); 64-wide j-panels of g double-buffered
// in LDS via async b128 (4 async instr / wave / panel).
// ---------------------------------------------------------------------------
#define SG_STR 80  // 320B row stride: 16B aligned; rows 2 apart -> 32 banks apart

__global__ __launch_bounds__(256) void attn_agg_kernel(
    const unsigned char* __restrict__ adj, const float* __restrict__ g,
    const float* __restrict__ el, const float* __restrict__ er,
    const float* __restrict__ mrow, const float* __restrict__ lrow,
    float* __restrict__ out) {
  __shared__ float sg[2][64][SG_STR];  // g panels (64 j x 64 f)
  __shared__ float ser[2][64];

  const int tid  = threadIdx.x;
  const int lane = tid & 31;
  const int wave = tid >> 5;
  const int hh = blockIdx.y;
  const int i0 = blockIdx.x * 128 + wave * 16;
  const int r    = lane & 15;
  const int half = lane >> 4;
  const int irow = i0 + r;

  const float el_r = el[irow * HEADS + hh];
  const float m_r  = mrow[irow * HEADS + hh];
  const float li_r = lrow[irow * HEADS + hh];

  v8f acc0 = {}, acc1 = {}, acc2 = {}, acc3 = {};

  auto stage = [&](int j0, int b) {
    #pragma unroll
    for (int e = 0; e < 4; ++e) {
      int ch = e * 256 + tid;               // 1024 chunks of 4 floats (64x64)
      int jj = ch >> 4, ff = (ch & 15) << 2;
      async_cp16(&g[(size_t)(j0 + jj) * OUTF + hh * HID + ff], &sg[b][jj][ff]);
    }
    if (tid < 64) ser[b][tid] = er[(j0 + tid) * HEADS + hh];
  };

  stage(0, 0);
  const int NP = N / 64;  // 64 j-panels
  for (int p = 0; p < NP; ++p) {
    const int b = p & 1;
    const int j0 = p * 64;
    if (p + 1 < NP) { stage(j0 + 64, b ^ 1); wait_async_le4(); }
    else            { wait_async_0(); }
    __syncthreads();

    #pragma unroll 4
    for (int jc = 0; jc < 64; jc += 4) {
      const int jb = jc + half * 2;  // this lane's 2 K slots
      const unsigned short ab =
          *(const unsigned short*)(adj + (size_t)irow * N + j0 + jb);
      float e0 = el_r + ser[b][jb];
      float e1 = el_r + ser[b][jb + 1];
      e0 = e0 >= 0.f ? e0 : SLOPE * e0;
      e1 = e1 >= 0.f ? e1 : SLOPE * e1;
      v2f a;
      a.x = (ab & 0xff) ? __expf(e0 - m_r) * li_r : 0.f;
      a.y = (ab >> 8)   ? __expf(e1 - m_r) * li_r : 0.f;

      v2f bf;
      bf.x = sg[b][jb][0 * 16 + r]; bf.y = sg[b][jb + 1][0 * 16 + r];
      acc0 = wmma_f32(a, bf, acc0);
      bf.x = sg[b][jb][1 * 16 + r]; bf.y = sg[b][jb + 1][1 * 16 + r];
      acc1 = wmma_f32(a, bf, acc1);
      bf.x = sg[b][jb][2 * 16 + r]; bf.y = sg[b][jb + 1][2 * 16 + r];
      acc2 = wmma_f32(a, bf, acc2);
      bf.x = sg[b][jb][3 * 16 + r]; bf.y = sg[b][jb + 1][3 * 16 + r];
      acc3 = wmma_f32(a, bf, acc3);
    }
    __syncthreads();
  }

  #pragma unroll
  for (int v = 0; v < 8; ++v) {
    int row = i0 + v + 8 * half;
    size_t base = (size_t)row * OUTF + hh * HID;
    out[base + 0 * 16 + r] = acc0[v];
    out[base + 1 * 16 + r] = acc1[v];
    out[base + 2 * 16 + r] = acc2[v];
    out[base + 3 * 16 + r] = acc3[v];
  }
}

// ---------------------------------------------------------------------------
extern "C" void kernel_launch(void* const* d_in, const int* in_sizes, int n_in,
                              void* d_out, int out_size, void* d_ws, size_t ws_size,
                              hipStream_t stream) {
  const float*         h   = (const float*)d_in[0];
  const unsigned char* adj = (const unsigned char*)d_in[1];  // bool N*N*1
  const float*         W   = (const float*)d_in[2];
  const float*         a   = (const float*)d_in[3];
  float*               out = (float*)d_out;

  float* g  = (float*)d_ws;                 // N*OUTF        (8 MB)
  float* el = g  + (size_t)N * OUTF;        // N*HEADS
  float* er = el + (size_t)N * HEADS;       // N*HEADS
  float* mr = er + (size_t)N * HEADS;       // N*HEADS
  float* lr = mr + (size_t)N * HEADS;       // N*HEADS (stores 1/l)

  dim3 gg(N / 64, OUTF / 64);               // 64 x 8 blocks
  gemm_g_kernel<<<gg, 256, 0, stream>>>(h, W, g);

  elr_kernel<<<N / 8, 256, 0, stream>>>(g, a, el, er);

  stats_kernel<<<N / 8, 256, 0, stream>>>(adj, el, er, mr, lr);

  dim3 ga(N / 128, HEADS);                  // 32 x 8 blocks
  attn_agg_kernel<<<ga, 256, 0, stream>>>(adj, g, el, er, mr, lr, out);
}